// MultiModalLightGCN_14551349199470
// MI455X (gfx1250) — compile-verified
//
#include <hip/hip_runtime.h>

typedef __attribute__((ext_vector_type(2))) float v2f;
typedef __attribute__((ext_vector_type(8))) float v8f;

constexpr int NUM_USERS = 100000;
constexpr int NUM_ITEMS = 50000;
constexpr int N_NODES   = NUM_USERS + NUM_ITEMS;
constexpr int EMB       = 64;
constexpr int N_EDGES   = 2000000;

// ---------------------------------------------------------------------------
// Weight loaders for the K-padded fused matmuls.
// Users: K = 12 (numeric) + 3 (vector), padded to 16.
// Items: K = 5 (numeric) + 12 (vector), padded to 20.
// W matrices are stored (EMB, K) row-major; we need B[k][n] = W[n][k].
// ---------------------------------------------------------------------------
__device__ __forceinline__ float uw_load(const float* __restrict__ Wnum,
                                         const float* __restrict__ Wvec,
                                         int n, int k) {
  if (k < 12) return Wnum[n * 12 + k];
  if (k < 15) return Wvec[n * 3 + (k - 12)];
  return 0.0f;
}

__device__ __forceinline__ float iw_load(const float* __restrict__ Wnum,
                                         const float* __restrict__ Wvec,
                                         int n, int k) {
  if (k < 5)  return Wnum[n * 5 + k];
  if (k < 17) return Wvec[n * 12 + (k - 5)];
  return 0.0f;
}

// ---------------------------------------------------------------------------
// User fusion: x[m] = emb_w[m] + F[m,:15] @ Wc.T + color_emb[ci[m]] +
//                     size_emb[si[m]] + b_num + b_vec
// One wave32 per 16-row tile. A is 16x4 f32 (v2f per lane), C/D 16x16 f32
// (v8f per lane), ISA layout:
//   A: lanes 0-15 -> M=0..15, vgpr0={K0 | K2 for lanes 16-31}, vgpr1={K1|K3}
//   C: vgpr i, lanes 0-15 -> M=i, N=lane ; lanes 16-31 -> M=i+8, N=lane-16
// ---------------------------------------------------------------------------
__global__ void fuse_users_wmma(const float* __restrict__ feat,
                                const int*   __restrict__ cidx,
                                const int*   __restrict__ sidx,
                                const float* __restrict__ emb_w,
                                const float* __restrict__ Wnum,
                                const float* __restrict__ bnum,
                                const float* __restrict__ Wvec,
                                const float* __restrict__ bvec,
                                const float* __restrict__ cemb,
                                const float* __restrict__ semb,
                                float*       __restrict__ x) {
  const int wave = (int)((blockIdx.x * blockDim.x + threadIdx.x) >> 5);
  if (wave * 16 >= NUM_USERS) return;           // wave-uniform: EXEC stays all-1
  const int lane = threadIdx.x & 31;
  const int half = lane >> 4;
  const int r    = lane & 15;
  const int m0   = wave * 16;

  // Load A tile (16 rows x K=16, zero-padded past 15) as 4 k-steps of 16x4.
  v2f a[4];
#pragma unroll
  for (int ks = 0; ks < 4; ++ks) {
    const int k = ks * 4 + 2 * half;
    a[ks].x = (k     < 15) ? feat[(m0 + r) * 15 + k]     : 0.0f;
    a[ks].y = (k + 1 < 15) ? feat[(m0 + r) * 15 + k + 1] : 0.0f;
  }

#pragma unroll
  for (int nt = 0; nt < 4; ++nt) {
    const int n = nt * 16 + r;
    v8f c = {};
#pragma unroll
    for (int ks = 0; ks < 4; ++ks) {
      const int k = ks * 4 + 2 * half;
      v2f b;
      b.x = uw_load(Wnum, Wvec, n, k);
      b.y = uw_load(Wnum, Wvec, n, k + 1);
      c = __builtin_amdgcn_wmma_f32_16x16x4_f32(false, a[ks], false, b,
                                                (short)0, c, false, false);
    }
    const float bias = bnum[n] + bvec[n];
#pragma unroll
    for (int i = 0; i < 8; ++i) {
      const int m = m0 + i + 8 * half;
      const float base = emb_w[m * EMB + n]
                       + cemb[cidx[m] * EMB + n]
                       + semb[sidx[m] * EMB + n]
                       + bias;
      x[m * EMB + n] = c[i] + base;
    }
  }
}

// ---------------------------------------------------------------------------
// Item fusion: x[NUM_USERS+m] = emb_w[m] + F[m,:17] @ Wc.T + b_num + b_vec
// K padded to 20 -> 5 k-steps of 16x4.
// ---------------------------------------------------------------------------
__global__ void fuse_items_wmma(const float* __restrict__ feat,
                                const float* __restrict__ emb_w,
                                const float* __restrict__ Wnum,
                                const float* __restrict__ bnum,
                                const float* __restrict__ Wvec,
                                const float* __restrict__ bvec,
                                float*       __restrict__ x) {
  const int wave = (int)((blockIdx.x * blockDim.x + threadIdx.x) >> 5);
  if (wave * 16 >= NUM_ITEMS) return;
  const int lane = threadIdx.x & 31;
  const int half = lane >> 4;
  const int r    = lane & 15;
  const int m0   = wave * 16;

  v2f a[5];
#pragma unroll
  for (int ks = 0; ks < 5; ++ks) {
    const int k = ks * 4 + 2 * half;
    a[ks].x = (k     < 17) ? feat[(m0 + r) * 17 + k]     : 0.0f;
    a[ks].y = (k + 1 < 17) ? feat[(m0 + r) * 17 + k + 1] : 0.0f;
  }

#pragma unroll
  for (int nt = 0; nt < 4; ++nt) {
    const int n = nt * 16 + r;
    v8f c = {};
#pragma unroll
    for (int ks = 0; ks < 5; ++ks) {
      const int k = ks * 4 + 2 * half;
      v2f b;
      b.x = iw_load(Wnum, Wvec, n, k);
      b.y = iw_load(Wnum, Wvec, n, k + 1);
      c = __builtin_amdgcn_wmma_f32_16x16x4_f32(false, a[ks], false, b,
                                                (short)0, c, false, false);
    }
    const float bias = bnum[n] + bvec[n];
#pragma unroll
    for (int i = 0; i < 8; ++i) {
      const int m = m0 + i + 8 * half;
      x[(NUM_USERS + m) * EMB + n] = c[i] + emb_w[m * EMB + n] + bias;
    }
  }
}

// ---------------------------------------------------------------------------
// Graph-side kernels (L2-resident scatter/gather; 38.4 MB node matrix << 192MB)
// ---------------------------------------------------------------------------
__global__ void zero_f32(float* __restrict__ p, int n) {
  const int i = blockIdx.x * blockDim.x + threadIdx.x;
  if (i < n) p[i] = 0.0f;
}

__global__ void deg_count(const int* __restrict__ col, float* __restrict__ deg) {
  const int e = blockIdx.x * blockDim.x + threadIdx.x;
  if (e < N_EDGES) atomicAdd(&deg[col[e]], 1.0f);
}

__global__ void deg_inv_sqrt(float* __restrict__ deg) {
  const int i = blockIdx.x * blockDim.x + threadIdx.x;
  if (i < N_NODES) {
    const float d = deg[i];
    deg[i] = (d > 0.0f) ? __frsqrt_rn(fmaxf(d, 1.0f)) : 0.0f;
  }
}

__global__ void edge_norm(const int* __restrict__ row, const int* __restrict__ col,
                          const float* __restrict__ dinv, float* __restrict__ norm) {
  const int e = blockIdx.x * blockDim.x + threadIdx.x;
  if (e < N_EDGES) norm[e] = dinv[row[e]] * dinv[col[e]];
}

// One edge handled by 16 lanes; each lane moves a float4 (256B/edge total).
__global__ void lightgcn_conv(const float* __restrict__ xin,
                              const int*   __restrict__ row,
                              const int*   __restrict__ col,
                              const float* __restrict__ norm,
                              float*       __restrict__ xout) {
  const int t = blockIdx.x * blockDim.x + threadIdx.x;
  const int e = t >> 4;
  if (e >= N_EDGES) return;
  const int d   = (t & 15) * 4;
  const int src = row[e];
  const int dst = col[e];
  const float nm = norm[e];
  const float4 v = *(const float4*)(xin + src * EMB + d);
  float* o = xout + dst * EMB + d;
  atomicAdd(o + 0, nm * v.x);
  atomicAdd(o + 1, nm * v.y);
  atomicAdd(o + 2, nm * v.z);
  atomicAdd(o + 3, nm * v.w);
}

__global__ void copy_f32(const float* __restrict__ src, float* __restrict__ dst, int n) {
  const int i = blockIdx.x * blockDim.x + threadIdx.x;
  if (i < n) dst[i] = src[i];
}

__global__ void add_f32(float* __restrict__ dst, const float* __restrict__ src, int n) {
  const int i = blockIdx.x * blockDim.x + threadIdx.x;
  if (i < n) dst[i] += src[i];
}

__global__ void scale_f32(float* __restrict__ p, float s, int n) {
  const int i = blockIdx.x * blockDim.x + threadIdx.x;
  if (i < n) p[i] *= s;
}

// ---------------------------------------------------------------------------
extern "C" void kernel_launch(void* const* d_in, const int* in_sizes, int n_in,
                              void* d_out, int out_size, void* d_ws, size_t ws_size,
                              hipStream_t stream) {
  (void)in_sizes; (void)n_in; (void)out_size; (void)ws_size;

  const int*   edge_index = (const int*)  d_in[0];   // [2, E]
  const float* user_feat  = (const float*)d_in[1];   // [100000, 15]
  const int*   ucolor     = (const int*)  d_in[2];   // [100000]
  const int*   usize      = (const int*)  d_in[3];   // [100000]
  const float* item_feat  = (const float*)d_in[4];   // [50000, 17]
  const float* user_emb_w = (const float*)d_in[5];   // [100000, 64]
  const float* item_emb_w = (const float*)d_in[6];   // [50000, 64]
  const float* u_num_W    = (const float*)d_in[7];   // [64, 12]
  const float* u_num_b    = (const float*)d_in[8];   // [64]
  const float* u_vec_W    = (const float*)d_in[9];   // [64, 3]
  const float* u_vec_b    = (const float*)d_in[10];  // [64]
  const float* color_emb  = (const float*)d_in[11];  // [22, 64]
  const float* size_emb   = (const float*)d_in[12];  // [18, 64]
  const float* i_num_W    = (const float*)d_in[13];  // [64, 5]
  const float* i_num_b    = (const float*)d_in[14];  // [64]
  const float* i_vec_W    = (const float*)d_in[15];  // [64, 12]
  const float* i_vec_b    = (const float*)d_in[16];  // [64]

  const int* row = edge_index;              // edge_index[0]
  const int* col = edge_index + N_EDGES;    // edge_index[1]

  float* out = (float*)d_out;               // [N_NODES, 64] running total

  // Workspace partition (~124 MB)
  float* x    = (float*)d_ws;               // [N_NODES*64]
  float* accA = x    + (size_t)N_NODES * EMB;
  float* accB = accA + (size_t)N_NODES * EMB;
  float* dinv = accB + (size_t)N_NODES * EMB;  // deg, then deg^-1/2 in place
  float* norm = dinv + N_NODES;                // [E]

  const int NV = N_NODES * EMB;             // 9,600,000

  // 1) Multimodal fusion (WMMA f32 16x16x4)
  {
    const int u_tiles = NUM_USERS / 16;     // 6250
    const int i_tiles = NUM_ITEMS / 16;     // 3125
    fuse_users_wmma<<<(u_tiles + 7) / 8, 256, 0, stream>>>(
        user_feat, ucolor, usize, user_emb_w,
        u_num_W, u_num_b, u_vec_W, u_vec_b, color_emb, size_emb, x);
    fuse_items_wmma<<<(i_tiles + 7) / 8, 256, 0, stream>>>(
        item_feat, item_emb_w, i_num_W, i_num_b, i_vec_W, i_vec_b, x);
  }

  // 2) Degrees and symmetric-normalization coefficients
  zero_f32<<<(N_NODES + 255) / 256, 256, 0, stream>>>(dinv, N_NODES);
  deg_count<<<(N_EDGES + 255) / 256, 256, 0, stream>>>(col, dinv);
  deg_inv_sqrt<<<(N_NODES + 255) / 256, 256, 0, stream>>>(dinv);
  edge_norm<<<(N_EDGES + 255) / 256, 256, 0, stream>>>(row, col, dinv, norm);

  // 3) total = x
  copy_f32<<<(NV + 255) / 256, 256, 0, stream>>>(x, out, NV);

  // 4) 3 propagation layers; ping-pong acc buffers, total accumulated in out
  const int conv_threads = N_EDGES * 16;
  const int conv_blocks  = (conv_threads + 255) / 256;

  zero_f32<<<(NV + 255) / 256, 256, 0, stream>>>(accA, NV);
  lightgcn_conv<<<conv_blocks, 256, 0, stream>>>(x, row, col, norm, accA);
  add_f32<<<(NV + 255) / 256, 256, 0, stream>>>(out, accA, NV);

  zero_f32<<<(NV + 255) / 256, 256, 0, stream>>>(accB, NV);
  lightgcn_conv<<<conv_blocks, 256, 0, stream>>>(accA, row, col, norm, accB);
  add_f32<<<(NV + 255) / 256, 256, 0, stream>>>(out, accB, NV);

  zero_f32<<<(NV + 255) / 256, 256, 0, stream>>>(accA, NV);
  lightgcn_conv<<<conv_blocks, 256, 0, stream>>>(accB, row, col, norm, accA);
  add_f32<<<(NV + 255) / 256, 256, 0, stream>>>(out, accA, NV);

  // 5) final = total / 4
  scale_f32<<<(NV + 255) / 256, 256, 0, stream>>>(out, 0.25f, NV);
}